// UnknownDetector_19653770346903
// MI455X (gfx1250) — compile-verified
//
#include <hip/hip_runtime.h>
#include <math.h>

// Problem constants (from reference): B=128, C=8, T=128, K=20, S=128
namespace {
constexpr int Bn = 128;
constexpr int Cn = 8;
constexpr int Tn = 128;
constexpr int Kn = 20;
constexpr int Sn = 128;
constexpr int WPB = 4;            // waves per block (one (b,k) DP per wave)
constexpr int ROWP = Sn + 4;      // LDS row pad: keeps 16B alignment, shifts banks
}

typedef float v2f __attribute__((ext_vector_type(2)));
typedef float v8f __attribute__((ext_vector_type(8)));

#define DTW_INF 3.0e37f

// ---- CDNA5 async (memory -> LDS) path, guarded so absence degrades cleanly ----
#if defined(__has_builtin)
#if __has_builtin(__builtin_amdgcn_global_load_async_to_lds_b128) && \
    __has_builtin(__builtin_amdgcn_global_load_async_to_lds_b32)
#define HAVE_ASYNC_LDS 1
#endif
#endif
#ifndef HAVE_ASYNC_LDS
#define HAVE_ASYNC_LDS 0
#endif

#if HAVE_ASYNC_LDS
// Exact parameter types per hipcc diagnostics:
//   b128: (int __vector(16B) AS(1)*, int __vector(16B) AS(3)*, Imm, Imm)
//   b32 : (int AS(1)*, int AS(3)*, Imm, Imm)
typedef int v4i32 __attribute__((vector_size(16)));
typedef __attribute__((address_space(1))) v4i32* gptr_b128;
typedef __attribute__((address_space(3))) v4i32* lptr_b128;
typedef __attribute__((address_space(1))) int*   gptr_b32;
typedef __attribute__((address_space(3))) int*   lptr_b32;
#endif

__device__ __forceinline__ void wait_async0() {
#if defined(__has_builtin) && __has_builtin(__builtin_amdgcn_s_wait_asynccnt)
  __builtin_amdgcn_s_wait_asynccnt(0);
#else
  asm volatile("s_wait_asynccnt 0x0" ::: "memory");
#endif
}

// One wave per (b,k): WMMA f32 16x16x4 cost tiles -> LDS -> tropical-scan DTW,
// then per-channel full cross-correlation max, both flags written to workspace.
__global__ __launch_bounds__(32 * WPB) void dtw_cc_kernel(
    const float* __restrict__ x,          // [B, C, T]
    const float* __restrict__ bary,       // [K, S, C]
    const float* __restrict__ top_dist,   // [K]
    const float* __restrict__ bottom_cc,  // [K, C]
    int* __restrict__ dtwF,               // [B*K]
    int* __restrict__ ccF)                // [B*K]
{
  __shared__ float lds_dot[WPB][16][ROWP];  // 16 x 128 dot strip per wave
  __shared__ float lds_x2[WPB][Tn];
  __shared__ float lds_xc[WPB][Tn];
  __shared__ float lds_vc[WPB][Tn];

  const int wave = threadIdx.x >> 5;
  const int lane = threadIdx.x & 31;
  const int bk   = blockIdx.x * WPB + wave;   // 640 blocks * 4 waves = 2560 exactly
  const int b    = bk / Kn;
  const int k    = bk - b * Kn;

  const float* __restrict__ xb = x + (size_t)b * (Cn * Tn);
  const float* __restrict__ vk = bary + (size_t)k * (Sn * Cn);

  __builtin_prefetch(xb, 0, 3);   // global_prefetch_b8: pull (b,k) slabs toward WGP
  __builtin_prefetch(vk, 0, 3);

  // ---- b2[s] for this lane's 4 columns (s = 4*lane + j), x2[t] -> LDS ----
  float b2l[4];
#pragma unroll
  for (int j = 0; j < 4; ++j) {
    const int s = 4 * lane + j;
    float acc = 0.f;
#pragma unroll
    for (int c = 0; c < Cn; ++c) { const float v = vk[s * Cn + c]; acc += v * v; }
    b2l[j] = acc;
  }
#pragma unroll
  for (int j = 0; j < 4; ++j) {
    const int t = 4 * lane + j;
    float acc = 0.f;
#pragma unroll
    for (int c = 0; c < Cn; ++c) { const float v = xb[c * Tn + t]; acc += v * v; }
    lds_x2[wave][t] = acc;
  }

  // ---- WMMA operand lane mapping (f32 16x16x4 layout, ISA 7.12.2) ----
  const int krow = (lane >> 4) * 2;  // K pair base: lanes 0-15 -> K{0,1}, 16-31 -> K{2,3}
  const int ncol = lane & 15;        // N / M-within-tile index
  const int rbase = 8 * (lane >> 4); // D tile: VGPR r holds M = r + 8*(lane>=16)

  // B operands reused across all 8 t-strips: B[c,s] = bary[k,s,c]
  v2f bmat[8][2];
#pragma unroll
  for (int sb = 0; sb < 8; ++sb) {
#pragma unroll
    for (int h = 0; h < 2; ++h) {
      const int c0 = h * 4 + krow;
      const int s  = sb * 16 + ncol;
      v2f t;
      t[0] = vk[s * Cn + c0 + 0];
      t[1] = vk[s * Cn + c0 + 1];
      bmat[sb][h] = t;
    }
  }

  float pv[4] = {0.f, 0.f, 0.f, 0.f};  // previous DTW row, 4 columns per lane

#pragma unroll 1
  for (int ts = 0; ts < 8; ++ts) {
    const int t0 = ts * 16;

    // A operands for this strip: A[t,c] = x[b,c,t]
    v2f amat[2];
#pragma unroll
    for (int h = 0; h < 2; ++h) {
      const int c0 = h * 4 + krow;
      v2f t;
      t[0] = xb[(c0 + 0) * Tn + t0 + ncol];
      t[1] = xb[(c0 + 1) * Tn + t0 + ncol];
      amat[h] = t;
    }

    // dots[t0:t0+16, 0:128] via 8 tiles x 2 chained K=4 WMMAs -> LDS strip
#pragma unroll
    for (int sb = 0; sb < 8; ++sb) {
      v8f d = {0.f, 0.f, 0.f, 0.f, 0.f, 0.f, 0.f, 0.f};
      d = __builtin_amdgcn_wmma_f32_16x16x4_f32(false, amat[0], false, bmat[sb][0],
                                                (short)0, d, false, false);
      d = __builtin_amdgcn_wmma_f32_16x16x4_f32(false, amat[1], false, bmat[sb][1],
                                                (short)0, d, false, false);
#pragma unroll
      for (int r = 0; r < 8; ++r)
        lds_dot[wave][rbase + r][sb * 16 + ncol] = d[r];
    }
    // Per-wave LDS slab; DS ops are in-order within a wave -> no barrier needed.

    // ---- DTW rows over this strip: tropical (min,+) scan per row ----
#pragma unroll 1
    for (int tt = 0; tt < 16; ++tt) {
      const int t = t0 + tt;
      const float4 dd = *(const float4*)&lds_dot[wave][tt][4 * lane];
      const float x2t = lds_x2[wave][t];
      const float c0 = fmaxf(x2t + b2l[0] - 2.f * dd.x, 0.f);
      const float c1 = fmaxf(x2t + b2l[1] - 2.f * dd.y, 0.f);
      const float c2 = fmaxf(x2t + b2l[2] - 2.f * dd.z, 0.f);
      const float c3 = fmaxf(x2t + b2l[3] - 2.f * dd.w, 0.f);

      if (t == 0) {
        // row 0: plain cumulative sum of cost
        const float s0 = c0, s1 = s0 + c1, s2 = s1 + c2, s3 = s2 + c3;
        float inc = s3;
#pragma unroll
        for (int off = 1; off < 32; off <<= 1) {
          const float o = __shfl_up(inc, off, 32);
          if (lane >= off) inc += o;
        }
        float excl = __shfl_up(inc, 1, 32);
        if (lane == 0) excl = 0.f;
        pv[0] = s0 + excl; pv[1] = s1 + excl; pv[2] = s2 + excl; pv[3] = s3 + excl;
      } else {
        // v[s] = min(b[s], v[s-1] + c[s]),  b[s] = c[s] + min(prev[s], prev[s-1])
        float plast = __shfl_up(pv[3], 1, 32);
        if (lane == 0) plast = DTW_INF;
        const float m0 = fminf(pv[0], plast);
        const float m1 = fminf(pv[1], pv[0]);
        const float m2 = fminf(pv[2], pv[1]);
        const float m3 = fminf(pv[3], pv[2]);
        const float b0 = c0 + m0, b1 = c1 + m1, b2 = c2 + m2, b3 = c3 + m3;

        // local composition of 4 affine min-plus maps: f(x)=min(Bc, x+Ac)
        float Ac = c0 + c1 + c2 + c3;
        float Bc = b0;
        Bc = fminf(b1, Bc + c1);
        Bc = fminf(b2, Bc + c2);
        Bc = fminf(b3, Bc + c3);

        // Hillis-Steele inclusive scan over lanes (non-commutative compose)
#pragma unroll
        for (int off = 1; off < 32; off <<= 1) {
          const float Ao = __shfl_up(Ac, off, 32);
          const float Bo = __shfl_up(Bc, off, 32);
          if (lane >= off) {
            Bc = fminf(Bc, Bo + Ac);
            Ac = Ac + Ao;
          }
        }
        float carry = __shfl_up(Bc, 1, 32);  // = v[4*lane - 1]
        if (lane == 0) carry = DTW_INF;
        const float v0 = fminf(b0, carry + c0);
        const float v1 = fminf(b1, v0 + c1);
        const float v2 = fminf(b2, v1 + c2);
        const float v3 = fminf(b3, v2 + c3);
        pv[0] = v0; pv[1] = v1; pv[2] = v2; pv[3] = v3;
      }
    }
  }

  const float dist = __shfl(pv[3], 31, 32);  // v[T-1][S-1]
  if (lane == 0) dtwF[bk] = (logf(dist) > top_dist[k]) ? 1 : 0;

  // ---- per-channel full cross-correlation max vs bottom_cc thresholds ----
  int fail = 0;
#pragma unroll 1
  for (int c = 0; c < Cn; ++c) {
#if HAVE_ASYNC_LDS
    // x[b,c,:] row is contiguous: one async b128 per lane straight into LDS
    __builtin_amdgcn_global_load_async_to_lds_b128(
        (gptr_b128)(xb + c * Tn + 4 * lane),
        (lptr_b128)&lds_xc[wave][4 * lane], 0, 0);
    // barycenter channel column (stride C): per-lane async b32 copies
#pragma unroll
    for (int j = 0; j < 4; ++j)
      __builtin_amdgcn_global_load_async_to_lds_b32(
          (gptr_b32)(vk + (4 * lane + j) * Cn + c),
          (lptr_b32)&lds_vc[wave][4 * lane + j], 0, 0);
    wait_async0();   // ASYNCcnt is not ordered with same-wave DS reads
#else
    const float4 xx = *(const float4*)&xb[c * Tn + 4 * lane];
    *(float4*)&lds_xc[wave][4 * lane] = xx;
#pragma unroll
    for (int j = 0; j < 4; ++j)
      lds_vc[wave][4 * lane + j] = vk[(4 * lane + j) * Cn + c];
    // same-wave LDS ordering -> no barrier
#endif

    float cmax = -DTW_INF;
#pragma unroll 1
    for (int r = 0; r < 8; ++r) {
      const int n = lane + 32 * r;           // 255 lags over 32 lanes
      if (n <= Tn + Sn - 2) {
        const int lag = n - (Sn - 1);        // lag in [-127, 127]
        const int tlo = lag > 0 ? lag : 0;
        const int thi = lag < 0 ? (Tn - 1 + lag) : (Tn - 1);
        float acc = 0.f;
        for (int t = tlo; t <= thi; ++t)
          acc = fmaf(lds_xc[wave][t], lds_vc[wave][t - lag], acc);
        cmax = fmaxf(cmax, acc);
      }
    }
#pragma unroll
    for (int off = 16; off >= 1; off >>= 1)
      cmax = fmaxf(cmax, __shfl_xor(cmax, off, 32));
    fail |= (cmax <= bottom_cc[k * Cn + c]) ? 1 : 0;
  }
  if (lane == 0) ccF[bk] = fail;
}

__global__ __launch_bounds__(Bn) void finalize_kernel(
    const int* __restrict__ preds,
    const int* __restrict__ dtwF,
    const int* __restrict__ ccF,
    int* __restrict__ out)
{
  const int b = threadIdx.x;
  if (b < Bn) {
    int dall = 1, call = 1;
#pragma unroll
    for (int k = 0; k < Kn; ++k) {
      dall &= dtwF[b * Kn + k];
      call &= ccF[b * Kn + k];
    }
    out[b] = (dall | call) ? Kn : preds[b];
  }
}

extern "C" void kernel_launch(void* const* d_in, const int* in_sizes, int n_in,
                              void* d_out, int out_size, void* d_ws, size_t ws_size,
                              hipStream_t stream) {
  (void)in_sizes; (void)n_in; (void)out_size; (void)ws_size;
  const float* x         = (const float*)d_in[0];
  const int*   preds     = (const int*)d_in[1];
  const float* bary      = (const float*)d_in[2];
  const float* top_dist  = (const float*)d_in[3];
  const float* bottom_cc = (const float*)d_in[4];
  int* out  = (int*)d_out;
  int* dtwF = (int*)d_ws;
  int* ccF  = dtwF + Bn * Kn;

  dtw_cc_kernel<<<(Bn * Kn) / WPB, 32 * WPB, 0, stream>>>(
      x, bary, top_dist, bottom_cc, dtwF, ccF);
  finalize_kernel<<<1, Bn, 0, stream>>>(preds, dtwF, ccF, out);
}